// TDSGCN_88699664597649
// MI455X (gfx1250) — compile-verified
//
#include <hip/hip_runtime.h>
#include <hip/hip_bf16.h>

// ---------------- problem constants (match reference) ----------------
#define N_USERS   4096
#define N_ITEMS   8192
#define N_NODES   (N_USERS + N_ITEMS)     // 12288
#define N_EDGES   393216
#define DIM       256
#define N_LAYERS  3
#define OUT_COLS  ((N_LAYERS + 1) * DIM)  // 1024
#define SLOPE     0.2f

typedef float v2f __attribute__((ext_vector_type(2)));
typedef float v8f __attribute__((ext_vector_type(8)));

// ---------------------------------------------------------------------
// Kernel 1: per-edge weight  ew[e] = vp[e]*w_p[src,dst] - vn[e]*w_n[src,dst]
// ---------------------------------------------------------------------
__global__ void edge_weight_kernel(const float* __restrict__ vp,
                                   const float* __restrict__ vn,
                                   const float* __restrict__ wp,
                                   const float* __restrict__ wn,
                                   const int*   __restrict__ src,
                                   const int*   __restrict__ dst,
                                   float* __restrict__ ew) {
    int e = blockIdx.x * blockDim.x + threadIdx.x;
    if (e >= N_EDGES) return;
    size_t idx = (size_t)src[e] * (size_t)N_NODES + (size_t)dst[e];
    ew[e] = vp[e] * wp[idx] - vn[e] * wn[idx];
}

// ---------------------------------------------------------------------
// Kernel 2: h0 = concat(user, item); also write layer-0 slice of output
// ---------------------------------------------------------------------
__global__ void init_h_kernel(const float* __restrict__ user,
                              const float* __restrict__ item,
                              float* __restrict__ h,
                              float* __restrict__ out) {
    size_t idx = (size_t)blockIdx.x * blockDim.x + threadIdx.x;  // node*DIM + d
    int node = (int)(idx >> 8);
    int d    = (int)(idx & 255);
    float v = (node < N_USERS) ? user[idx]
                               : item[idx - (size_t)N_USERS * DIM];
    h[idx] = v;
    out[(size_t)node * OUT_COLS + d] = v;
}

// ---------------------------------------------------------------------
// Kernel 3: hw = h @ W  via V_WMMA_F32_16X16X4_F32 (exact fp32 path).
// One wave32 computes one 16x16 tile of hw; K=256 in 64 WMMA steps.
//
// f32 operand layouts (CDNA5 ISA 7.12.2):
//   A 16x4 : lane L -> M = L&15 ; VGPR j -> K = 2*(L>>4) + j
//   B 4x16 : lane L -> N = L&15 ; VGPR j -> K = 2*(L>>4) + j
//   C 16x16: lane L -> N = L&15 ; VGPR r -> M = r + 8*(L>>4)
// ---------------------------------------------------------------------
#define GEMM_WAVES 4
__global__ void gemm_wmma_f32_kernel(const float* __restrict__ H,
                                     const float* __restrict__ W,
                                     float* __restrict__ HW) {
    const int lane = threadIdx.x & 31;
    const int wave = threadIdx.x >> 5;
    const int tile = blockIdx.x * GEMM_WAVES + wave;   // 12288 tiles total
    const int tilesN = DIM / 16;                       // 16
    const int tileM = tile / tilesN;                   // 0..767
    const int tileN = tile % tilesN;                   // 0..15
    const int row0 = tileM * 16;
    const int col0 = tileN * 16;

    const int m = lane & 15;        // M for A, N for B/C
    const int g = lane >> 4;        // lane-half

    v8f c = {0.f, 0.f, 0.f, 0.f, 0.f, 0.f, 0.f, 0.f};

    const float* __restrict__ arow = H + (size_t)(row0 + m) * DIM + 2 * g;
    const float* __restrict__ bcol = W + (size_t)(2 * g) * DIM + col0 + m;

    #pragma unroll 4
    for (int k = 0; k < DIM; k += 4) {
        v2f a, b;
        a.x = arow[k];           // K = k + 2g
        a.y = arow[k + 1];       // K = k + 2g + 1
        b.x = bcol[(size_t)k * DIM];           // W[k+2g,       col0+m]
        b.y = bcol[(size_t)(k + 1) * DIM];     // W[k+2g+1,     col0+m]
        c = __builtin_amdgcn_wmma_f32_16x16x4_f32(
                /*neg_a=*/false, a, /*neg_b=*/false, b,
                /*c_mod=*/(short)0, c, /*reuse_a=*/false, /*reuse_b=*/false);
    }

    float* __restrict__ outp = HW + (size_t)(row0 + 8 * g) * DIM + col0 + m;
    #pragma unroll
    for (int r = 0; r < 8; ++r)
        outp[(size_t)r * DIM] = c[r];
}

// ---------------------------------------------------------------------
// Kernel 4: edge scatter. One wave per edge; each lane owns 8 floats
// (two float4 chunks). agg[dst] += (hw[src] + time_table[t]) * ew[e]
// via hardware global_atomic_add_f32.
// ---------------------------------------------------------------------
#define SC_WAVES 8
__global__ void edge_scatter_kernel(const float* __restrict__ hw,
                                    const float* __restrict__ tt,
                                    const float* __restrict__ ew,
                                    const int*   __restrict__ src,
                                    const int*   __restrict__ dst,
                                    const int*   __restrict__ tix,
                                    float* __restrict__ agg) {
    const int e = blockIdx.x * SC_WAVES + (threadIdx.x >> 5);
    if (e >= N_EDGES) return;
    const int lane = threadIdx.x & 31;

    const int s = src[e];
    const int d = dst[e];
    const int t = tix[e];
    const float w = ew[e];

    const float4* __restrict__ hp = (const float4*)(hw + (size_t)s * DIM) + lane;
    const float4* __restrict__ tp = (const float4*)(tt + (size_t)t * DIM) + lane;
    float4 h0 = hp[0],  h1 = hp[32];
    float4 t0 = tp[0],  t1 = tp[32];

    float* __restrict__ ag = agg + (size_t)d * DIM + lane * 4;
    unsafeAtomicAdd(ag + 0,       (h0.x + t0.x) * w);
    unsafeAtomicAdd(ag + 1,       (h0.y + t0.y) * w);
    unsafeAtomicAdd(ag + 2,       (h0.z + t0.z) * w);
    unsafeAtomicAdd(ag + 3,       (h0.w + t0.w) * w);
    unsafeAtomicAdd(ag + 128 + 0, (h1.x + t1.x) * w);
    unsafeAtomicAdd(ag + 128 + 1, (h1.y + t1.y) * w);
    unsafeAtomicAdd(ag + 128 + 2, (h1.z + t1.z) * w);
    unsafeAtomicAdd(ag + 128 + 3, (h1.w + t1.w) * w);
}

// ---------------------------------------------------------------------
// Kernel 5: h = leaky_relu(agg); out slice = h / max(||h||_2, 1e-12)
// One 256-thread block per node row. wave32 shuffle reduction.
// ---------------------------------------------------------------------
__global__ void act_norm_kernel(const float* __restrict__ agg,
                                float* __restrict__ h,
                                float* __restrict__ out,
                                int layer) {
    const int node = blockIdx.x;
    const int d    = threadIdx.x;          // 0..255
    const size_t base = (size_t)node * DIM;

    float x = agg[base + d];
    x = (x > 0.f) ? x : SLOPE * x;
    h[base + d] = x;

    // block reduction of x*x across 8 wave32s
    float ss = x * x;
    #pragma unroll
    for (int off = 16; off > 0; off >>= 1)
        ss += __shfl_xor(ss, off, 32);

    __shared__ float red[8];
    const int lane = threadIdx.x & 31;
    const int wv   = threadIdx.x >> 5;
    if (lane == 0) red[wv] = ss;
    __syncthreads();
    if (wv == 0) {
        float v = (lane < 8) ? red[lane] : 0.f;
        #pragma unroll
        for (int off = 4; off > 0; off >>= 1)
            v += __shfl_xor(v, off, 32);
        if (lane == 0) red[0] = v;
    }
    __syncthreads();

    float nrm = fmaxf(sqrtf(red[0]), 1e-12f);
    out[(size_t)node * OUT_COLS + (size_t)(layer + 1) * DIM + d] = x / nrm;
}

// ---------------------------------------------------------------------
// Host-side orchestration
// ---------------------------------------------------------------------
extern "C" void kernel_launch(void* const* d_in, const int* in_sizes, int n_in,
                              void* d_out, int out_size, void* d_ws, size_t ws_size,
                              hipStream_t stream) {
    const float* user_embd  = (const float*)d_in[0];
    const float* item_embd  = (const float*)d_in[1];
    const float* w_p        = (const float*)d_in[2];
    const float* w_n        = (const float*)d_in[3];
    const float* layer_W    = (const float*)d_in[4];   // [3,256,256]
    const float* time_table = (const float*)d_in[5];   // [366,256]
    const float* val_p      = (const float*)d_in[6];
    const float* val_n      = (const float*)d_in[7];
    const int*   src        = (const int*)d_in[8];
    const int*   dst        = (const int*)d_in[9];
    const int*   time_idx   = (const int*)d_in[10];
    float* out = (float*)d_out;

    // workspace layout
    const size_t nodeBytes = (size_t)N_NODES * DIM * sizeof(float);  // 12.58 MB
    char* ws = (char*)d_ws;
    float* h    = (float*)(ws);
    float* hw   = (float*)(ws + nodeBytes);
    float* agg  = (float*)(ws + 2 * nodeBytes);
    float* ew   = (float*)(ws + 3 * nodeBytes);                      // E floats

    // 1) edge weights
    edge_weight_kernel<<<(N_EDGES + 255) / 256, 256, 0, stream>>>(
        val_p, val_n, w_p, w_n, src, dst, ew);

    // 2) h0 = concat(user, item); write layer-0 output slice
    {
        const size_t total = (size_t)N_NODES * DIM;
        init_h_kernel<<<(unsigned)(total / 256), 256, 0, stream>>>(
            user_embd, item_embd, h, out);
    }

    // 3) three GCN layers
    const int gemmTiles  = (N_NODES / 16) * (DIM / 16);          // 12288 tiles
    const int gemmBlocks = gemmTiles / GEMM_WAVES;               // 3072
    const int scBlocks   = N_EDGES / SC_WAVES;                   // 49152

    for (int i = 0; i < N_LAYERS; ++i) {
        const float* Wi = layer_W + (size_t)i * DIM * DIM;

        gemm_wmma_f32_kernel<<<gemmBlocks, 32 * GEMM_WAVES, 0, stream>>>(h, Wi, hw);

        hipMemsetAsync(agg, 0, nodeBytes, stream);

        edge_scatter_kernel<<<scBlocks, 32 * SC_WAVES, 0, stream>>>(
            hw, time_table, ew, src, dst, time_idx, agg);

        act_norm_kernel<<<N_NODES, DIM, 0, stream>>>(agg, h, out, i);
    }
}